// TemporalGCN_44178033607092
// MI455X (gfx1250) — compile-verified
//
#include <hip/hip_runtime.h>
#include <hip/hip_bf16.h>

#define NN 10000
#define EE 80000
#define BB 16
#define HH 64
#define TT 12
#define BN (BB*NN)          // 160000 node-rows
#define RST 68              // padded LDS row stride (conflict-free b64 frags)
#define WPB 8               // waves per block (256 threads)
#define ROWS_PER_BLOCK (WPB*16)

typedef __attribute__((ext_vector_type(2))) float v2f;
typedef __attribute__((ext_vector_type(8))) float v8f;

// Fast activations built on the hardware transcendental (v_exp_f32 / v_rcp_f32).
// Both are saturation-safe at the extremes (no inf*0 NaNs).
__device__ __forceinline__ float fast_sigmoid(float x) {
    return __builtin_amdgcn_rcpf(1.0f + __expf(-x));
}
__device__ __forceinline__ float fast_tanh(float x) {
    float e = __expf(-2.0f * __builtin_fabsf(x));      // e in (0,1]
    float r = (1.0f - e) * __builtin_amdgcn_rcpf(1.0f + e);
    return __builtin_copysignf(r, x);
}

// ---------------- graph normalization ----------------
__global__ void k_init_deg(float* __restrict__ deg) {
    int i = blockIdx.x * 256 + threadIdx.x;
    if (i < NN) deg[i] = 2.0f;                       // improved self-loop fill
}
__global__ void k_acc_deg(const int* __restrict__ ei, const float* __restrict__ ew,
                          float* __restrict__ deg) {
    int e = blockIdx.x * 256 + threadIdx.x;
    if (e < EE) atomicAdd(&deg[ei[EE + e]], ew[e]);  // segment_sum over col
}
__global__ void k_dinv(const float* __restrict__ deg, float* __restrict__ dinv) {
    int i = blockIdx.x * 256 + threadIdx.x;
    if (i < NN) { float d = deg[i]; dinv[i] = (d > 0.0f) ? rsqrtf(d) : 0.0f; }
}
__global__ void k_norm(const int* __restrict__ ei, const float* __restrict__ ew,
                       const float* __restrict__ dinv, float* __restrict__ nrm) {
    int e = blockIdx.x * 256 + threadIdx.x;
    if (e < EE) nrm[e] = dinv[ei[e]] * ew[e] * dinv[ei[EE + e]];
}

// ---------------- feature aggregation: A[t, b*N+n, f] ----------------
__global__ void k_initA(const float* __restrict__ x, const float* __restrict__ dinv,
                        float* __restrict__ A) {
    int i = blockIdx.x * 256 + threadIdx.x;          // over BN*24 (x-contiguous)
    if (i < BN * 24) {
        int bn = i / 24, k = i - bn * 24;
        int f = k / 12, t = k - f * 12;
        int node = bn % NN;
        float dv = dinv[node];
        A[((size_t)t * BN + bn) * 2 + f] = 2.0f * dv * dv * x[i];
    }
}
__global__ void k_scatter(const int* __restrict__ ei, const float* __restrict__ nrm,
                          const float* __restrict__ x, float* __restrict__ A) {
    long i = (long)blockIdx.x * 256 + threadIdx.x;   // over EE*BB*24
    if (i < (long)EE * BB * 24) {
        int k = (int)(i % 24); long r = i / 24;
        int b = (int)(r % BB); int e = (int)(r / BB);
        int row = ei[e], col = ei[EE + e];
        int f = k / 12, t = k - f * 12;
        float v = nrm[e] * x[(size_t)(b * NN + row) * 24 + k];
        atomicAdd(&A[((size_t)t * BN + (size_t)(b * NN + col)) * 2 + f], v);
    }
}

// ---------------- fused 12-step GRU scan with fp32 WMMA ----------------
#define GATE_MATMUL(G, ACC)                                                     \
    _Pragma("unroll")                                                           \
    for (int j = 0; j < 4; ++j) {                                               \
        const int col = j * 16 + n;                                             \
        const float c0 = Cg[(G) * 64 + col];                                    \
        const float m0 = Mg[(G) * 128 + col];                                   \
        const float m1 = Mg[(G) * 128 + 64 + col];                              \
        v8f acc;                                                                \
        _Pragma("unroll")                                                       \
        for (int v = 0; v < 8; ++v) acc[v] = c0 + m0 * a0v[v] + m1 * a1v[v];    \
        const float* Ug = &Ut[(G) * 64 * RST + col * RST];                      \
        _Pragma("unroll")                                                       \
        for (int k = 0; k < 16; ++k) {                                          \
            v2f af = *(const v2f*)&Htile[n * RST + 4 * k + kb];                 \
            v2f bf = *(const v2f*)&Ug[4 * k + kb];                              \
            acc = __builtin_amdgcn_wmma_f32_16x16x4_f32(                        \
                false, af, false, bf, (short)0, acc, false, false);             \
        }                                                                       \
        ACC[j] = acc;                                                           \
    }

__global__ __launch_bounds__(256)
void k_scan(const float* __restrict__ A,
            const float* __restrict__ Wz, const float* __restrict__ bz,
            const float* __restrict__ Wr, const float* __restrict__ br,
            const float* __restrict__ Wh, const float* __restrict__ bh,
            const float* __restrict__ LzW, const float* __restrict__ Lzb,
            const float* __restrict__ LrW, const float* __restrict__ Lrb,
            const float* __restrict__ LhW, const float* __restrict__ Lhb,
            const float* __restrict__ Wout, const float* __restrict__ bout,
            float* __restrict__ out) {
    extern __shared__ float smem[];
    float* Ut = smem;                       // 3 * 64*RST  (U^T, padded rows)
    float* Hl = Ut + 3 * 64 * RST;          // WPB * 16*RST (per-wave H tiles)
    float* Mg = Hl + WPB * 16 * RST;        // 3 * 128 (folded A-projections)
    float* Cg = Mg + 3 * 128;               // 3 * 64  (folded biases)
    float* Aw = Cg + 3 * 64;                // WPB * 32 (A staging)

    const int tid  = threadIdx.x;
    const int w    = tid >> 5;
    const int lane = tid & 31;
    const int n    = lane & 15;             // col-in-tile == A-frag row
    const int hi   = lane >> 4;
    const int kb   = hi << 1;               // K sub-pair select

    // --- prologue: build U^T tiles and folded M/c (L2-cached reads) ---
    for (int idx = tid; idx < 3 * 64 * 64; idx += 256) {
        int g = idx >> 12, r = idx & 4095, nn = r >> 6, k = r & 63;
        const float* LW = (g == 0) ? LzW : (g == 1) ? LrW : LhW;
        Ut[g * 64 * RST + nn * RST + k] = LW[(64 + k) * 64 + nn];  // U^T[n][k]
    }
    for (int idx = tid; idx < 3 * 3 * 64; idx += 256) {
        int g = idx / 192, r = idx - g * 192, rt = r >> 6, col = r & 63;
        const float* LW = (g == 0) ? LzW : (g == 1) ? LrW : LhW;
        float s = 0.0f;
        if (rt < 2) {
            const float* Wg = (g == 0) ? Wz : (g == 1) ? Wr : Wh;
            for (int h = 0; h < 64; ++h) s += Wg[rt * 64 + h] * LW[h * 64 + col];
            Mg[g * 128 + rt * 64 + col] = s;
        } else {
            const float* bg = (g == 0) ? bz : (g == 1) ? br : bh;
            const float* Lb = (g == 0) ? Lzb : (g == 1) ? Lrb : Lhb;
            for (int h = 0; h < 64; ++h) s += bg[h] * LW[h * 64 + col];
            Cg[g * 64 + col] = s + Lb[col];
        }
    }
    for (int idx = tid; idx < WPB * 16 * RST; idx += 256) Hl[idx] = 0.0f; // H0 = 0
    __syncthreads();

    const int row_base = blockIdx.x * ROWS_PER_BLOCK + w * 16;
    float* Htile = &Hl[w * 16 * RST];

    v8f hreg[4];
    #pragma unroll
    for (int j = 0; j < 4; ++j)
        #pragma unroll
        for (int v = 0; v < 8; ++v) hreg[j][v] = 0.0f;

    for (int t = 0; t < TT; ++t) {
        // stage this wave's 16 rows of aggregated features (coalesced)
        Aw[w * 32 + lane] = A[((size_t)t * BN + row_base) * 2 + lane];
        float a0v[8], a1v[8];
        #pragma unroll
        for (int v = 0; v < 8; ++v) {
            int m = hi * 8 + v;
            a0v[v] = Aw[w * 32 + 2 * m];
            a1v[v] = Aw[w * 32 + 2 * m + 1];
        }

        v8f zacc[4], racc[4];
        GATE_MATMUL(0, zacc)                 // Z pre-activation = A·Mz + H·Uz + cz
        GATE_MATMUL(1, racc)                 // R pre-activation

        // activations; overwrite LDS H tile with H*R (wave-local, DS in-order)
        #pragma unroll
        for (int j = 0; j < 4; ++j)
            #pragma unroll
            for (int v = 0; v < 8; ++v) {
                float z = fast_sigmoid(zacc[j][v]);
                float r = fast_sigmoid(racc[j][v]);
                zacc[j][v] = z;
                int m = hi * 8 + v;
                Htile[m * RST + j * 16 + n] = hreg[j][v] * r;
            }

        v8f hacc[4];
        GATE_MATMUL(2, hacc)                 // candidate: A·Mh + (H∘R)·Uh + ch

        #pragma unroll
        for (int j = 0; j < 4; ++j)
            #pragma unroll
            for (int v = 0; v < 8; ++v) {
                float ht = fast_tanh(hacc[j][v]);
                float z  = zacc[j][v];
                float hn = z * hreg[j][v] + (1.0f - z) * ht;
                hreg[j][v] = hn;
                int m = hi * 8 + v;
                Htile[m * RST + j * 16 + n] = hn;   // restore H for next step
            }
    }

    // --- epilogue: H out, then y = exp(H @ Wout + bout) ---
    #pragma unroll
    for (int j = 0; j < 4; ++j)
        #pragma unroll
        for (int v = 0; v < 8; ++v) {
            int m = hi * 8 + v;
            out[(size_t)BN + (size_t)(row_base + m) * HH + j * 16 + n] = hreg[j][v];
        }
    if (hi == 0) {
        float s = 0.0f;
        for (int k = 0; k < 64; ++k) s += Htile[n * RST + k] * Wout[k];
        out[row_base + n] = expf(s + bout[0]);   // precise exp for final output
    }
}

extern "C" void kernel_launch(void* const* d_in, const int* in_sizes, int n_in,
                              void* d_out, int out_size, void* d_ws, size_t ws_size,
                              hipStream_t stream) {
    const float* x    = (const float*)d_in[0];
    const int*   ei   = (const int*)  d_in[1];
    const float* ew   = (const float*)d_in[2];
    const float* Wz   = (const float*)d_in[3];
    const float* bz   = (const float*)d_in[4];
    const float* Wr   = (const float*)d_in[5];
    const float* br   = (const float*)d_in[6];
    const float* Wh   = (const float*)d_in[7];
    const float* bh   = (const float*)d_in[8];
    const float* LzW  = (const float*)d_in[9];
    const float* Lzb  = (const float*)d_in[10];
    const float* LrW  = (const float*)d_in[11];
    const float* Lrb  = (const float*)d_in[12];
    const float* LhW  = (const float*)d_in[13];
    const float* Lhb  = (const float*)d_in[14];
    const float* Wout = (const float*)d_in[15];
    const float* bout = (const float*)d_in[16];
    float* out = (float*)d_out;

    float* deg  = (float*)d_ws;        // N
    float* dinv = deg + NN;            // N
    float* nrm  = dinv + NN;           // E
    float* Abuf = nrm + EE;            // T*BN*2 (~15.4 MB)

    k_init_deg<<<(NN + 255) / 256, 256, 0, stream>>>(deg);
    k_acc_deg <<<(EE + 255) / 256, 256, 0, stream>>>(ei, ew, deg);
    k_dinv    <<<(NN + 255) / 256, 256, 0, stream>>>(deg, dinv);
    k_norm    <<<(EE + 255) / 256, 256, 0, stream>>>(ei, ew, dinv, nrm);
    k_initA   <<<(BN * 24 + 255) / 256, 256, 0, stream>>>(x, dinv, Abuf);
    long total = (long)EE * BB * 24;
    k_scatter <<<(int)((total + 255) / 256), 256, 0, stream>>>(ei, nrm, x, Abuf);

    size_t smem = (size_t)(3 * 64 * RST + WPB * 16 * RST + 3 * 128 + 3 * 64 + WPB * 32)
                  * sizeof(float);     // ~90 KB (CDNA5 WGP allows up to 320 KB)
    k_scan<<<BN / ROWS_PER_BLOCK, 256, smem, stream>>>(
        Abuf, Wz, bz, Wr, br, Wh, bh, LzW, Lzb, LrW, Lrb, LhW, Lhb, Wout, bout, out);
}